// Text_GCN_9371618640020
// MI455X (gfx1250) — compile-verified
//
#include <hip/hip_runtime.h>
#include <hip/hip_bf16.h>

typedef float v2f __attribute__((ext_vector_type(2)));
typedef float v8f __attribute__((ext_vector_type(8)));

#define GCN_N 50000
#define GCN_INC 512
#define GCN_HID 128
#define GCN_OUT 64

// ---------------------------------------------------------------------------
// Degree / normalization
// ---------------------------------------------------------------------------
__global__ void k_deg_init(float* deg, int n) {
    int i = blockIdx.x * blockDim.x + threadIdx.x;
    if (i < n) deg[i] = 1.0f;  // self-loop contributes weight 1 to deg[i]
}

__global__ void k_deg_edges(const long long* __restrict__ ei,
                            const float* __restrict__ ew,
                            float* deg, int e_cnt) {
    int e = blockIdx.x * blockDim.x + threadIdx.x;
    if (e < e_cnt) {
        long long c = ei[(size_t)e_cnt + e];  // col = edge_index[1][e]
        atomicAdd(&deg[c], ew[e]);
    }
}

__global__ void k_dinv(float* deg, int n) {
    int i = blockIdx.x * blockDim.x + threadIdx.x;
    if (i < n) {
        float d = deg[i];
        deg[i] = (d > 0.0f) ? rsqrtf(d) : 0.0f;  // d >= 1 always here
    }
}

// ---------------------------------------------------------------------------
// WMMA f32 GEMM: C[M,NDIM] = A[M,KDIM] * B[KDIM,NDIM]
// One wave computes a 16 x NDIM stripe using V_WMMA_F32_16X16X4_F32.
// A 16x4 f32 layout: lanes 0-15 -> M=lane, v0=K0 v1=K1; lanes 16-31 -> K2,K3.
// B 4x16 f32 layout: lanes 0-15 -> N=lane, v0=K0 v1=K1; lanes 16-31 -> K2,K3.
// C/D 16x16 f32: VGPR r -> M=r (lanes 0-15), M=8+r (lanes 16-31), N=lane&15.
// ---------------------------------------------------------------------------
template <int KDIM, int NDIM>
__global__ void k_gemm_wmma(const float* __restrict__ A,
                            const float* __restrict__ B,
                            float* __restrict__ C, int M) {
    constexpr int NT = NDIM / 16;                 // N tiles per wave
    const int wave_in_blk = threadIdx.x >> 5;
    const int mt = blockIdx.x * (blockDim.x >> 5) + wave_in_blk;
    if (mt * 16 >= M) return;                     // wave-uniform exit
    const int lane = threadIdx.x & 31;
    const int half = lane >> 4;
    const int l = lane & 15;

    v8f acc[NT];
#pragma unroll
    for (int j = 0; j < NT; ++j)
#pragma unroll
        for (int r = 0; r < 8; ++r) acc[j][r] = 0.0f;

    const float* arow = A + (size_t)(mt * 16 + l) * KDIM + 2 * half;
    for (int k = 0; k < KDIM; k += 4) {
        v2f a;
        a.x = arow[k];
        a.y = arow[k + 1];
#pragma unroll
        for (int j = 0; j < NT; ++j) {
            const float* bp = B + (size_t)(k + 2 * half) * NDIM + j * 16 + l;
            v2f b;
            b.x = bp[0];
            b.y = bp[NDIM];
            acc[j] = __builtin_amdgcn_wmma_f32_16x16x4_f32(
                false, a, false, b, (short)0, acc[j], false, false);
        }
    }

#pragma unroll
    for (int j = 0; j < NT; ++j) {
        float* crow = C + (size_t)(mt * 16 + half * 8) * NDIM + j * 16 + l;
#pragma unroll
        for (int r = 0; r < 8; ++r) crow[(size_t)r * NDIM] = acc[j][r];
    }
}

// ---------------------------------------------------------------------------
// Self-loop initialization: dst[i,f] = dinv[i]^2 * src[i,f] (+ optional bias)
// ---------------------------------------------------------------------------
template <int F, bool WITH_BIAS>
__global__ void k_selfloop_init(const float* __restrict__ src,
                                const float* __restrict__ dinv,
                                const float* __restrict__ bias,
                                float* __restrict__ dst, int n) {
    int t = blockIdx.x * blockDim.x + threadIdx.x;
    if (t >= n * F) return;
    int i = t / F;
    int f = t - i * F;
    float di = dinv[i];
    float v = di * di * src[t];
    if (WITH_BIAS) v += bias[f];
    dst[t] = v;
}

// ---------------------------------------------------------------------------
// Edge scatter: one wave per edge; lane handles 4 features via float4 gather.
// dst[col,f] += dinv[row]*ew*dinv[col] * src[row,f]
// ---------------------------------------------------------------------------
template <int F>
__global__ void k_scatter_edges(const long long* __restrict__ ei,
                                const float* __restrict__ ew,
                                const float* __restrict__ dinv,
                                const float* __restrict__ src,
                                float* __restrict__ dst, int e_cnt) {
    int e = blockIdx.x * (blockDim.x >> 5) + (threadIdx.x >> 5);
    if (e >= e_cnt) return;
    int lane = threadIdx.x & 31;
    long long r = ei[e];
    long long c = ei[(size_t)e_cnt + e];
    float nrm = dinv[r] * ew[e] * dinv[c];
    const float4* s = (const float4*)(src + (size_t)r * F);
    float* d = dst + (size_t)c * F;
#pragma unroll 1
    for (int f4 = lane; f4 < F / 4; f4 += 32) {
        float4 v = s[f4];
        atomicAdd(d + 4 * f4 + 0, nrm * v.x);
        atomicAdd(d + 4 * f4 + 1, nrm * v.y);
        atomicAdd(d + 4 * f4 + 2, nrm * v.z);
        atomicAdd(d + 4 * f4 + 3, nrm * v.w);
    }
}

// ---------------------------------------------------------------------------
// Bias + ReLU in place: h[i,f] = max(h[i,f] + b[f], 0)
// ---------------------------------------------------------------------------
template <int F>
__global__ void k_bias_relu(float* __restrict__ h, const float* __restrict__ b,
                            int n) {
    int t = blockIdx.x * blockDim.x + threadIdx.x;
    if (t >= n * F) return;
    int f = t % F;
    float v = h[t] + b[f];
    h[t] = v > 0.0f ? v : 0.0f;
}

// ---------------------------------------------------------------------------
extern "C" void kernel_launch(void* const* d_in, const int* in_sizes, int n_in,
                              void* d_out, int out_size, void* d_ws, size_t ws_size,
                              hipStream_t stream) {
    const float* x       = (const float*)d_in[0];       // [N, 512]
    const long long* ei  = (const long long*)d_in[1];   // [2, E] int64
    const float* ew      = (const float*)d_in[2];       // [E]
    const float* W1      = (const float*)d_in[3];       // [512, 128]
    const float* b1      = (const float*)d_in[4];       // [128]
    const float* W2      = (const float*)d_in[5];       // [128, 64]
    const float* b2      = (const float*)d_in[6];       // [64]
    float* out           = (float*)d_out;               // [N, 64]

    const int N = in_sizes[0] / GCN_INC;
    const int E = in_sizes[2];

    // Workspace layout (bytes)
    char* ws = (char*)d_ws;
    size_t off = 0;
    float* dinv = (float*)(ws + off);  off += (((size_t)N * 4 + 255) / 256) * 256;
    float* xw1  = (float*)(ws + off);  off += (size_t)N * GCN_HID * 4;
    float* h1   = (float*)(ws + off);  off += (size_t)N * GCN_HID * 4;
    float* xw2  = (float*)(ws + off);  off += (size_t)N * GCN_OUT * 4;
    (void)ws_size;

    const int TPB = 256;

    // --- normalization: deg -> dinv (in place) ---
    k_deg_init<<<(N + TPB - 1) / TPB, TPB, 0, stream>>>(dinv, N);
    k_deg_edges<<<(E + TPB - 1) / TPB, TPB, 0, stream>>>(ei, ew, dinv, E);
    k_dinv<<<(N + TPB - 1) / TPB, TPB, 0, stream>>>(dinv, N);

    // --- layer 1: xw1 = x @ W1 (WMMA f32) ---
    {
        int mtiles = (N + 15) / 16;             // 3125
        int wpb = 4;                            // 4 waves / block
        int blocks = (mtiles + wpb - 1) / wpb;
        k_gemm_wmma<GCN_INC, GCN_HID><<<blocks, wpb * 32, 0, stream>>>(x, W1, xw1, N);
    }
    // self-loop term initializes accumulator (no zeroing, no atomics for loops)
    k_selfloop_init<GCN_HID, false><<<((size_t)N * GCN_HID + TPB - 1) / TPB, TPB, 0, stream>>>(
        xw1, dinv, nullptr, h1, N);
    // real edges: wave per edge, 4 edges per 128-thread block
    k_scatter_edges<GCN_HID><<<(E + 3) / 4, 128, 0, stream>>>(ei, ew, dinv, xw1, h1, E);
    k_bias_relu<GCN_HID><<<((size_t)N * GCN_HID + TPB - 1) / TPB, TPB, 0, stream>>>(h1, b1, N);

    // --- layer 2: xw2 = h1 @ W2 (WMMA f32) ---
    {
        int mtiles = (N + 15) / 16;
        int wpb = 4;
        int blocks = (mtiles + wpb - 1) / wpb;
        k_gemm_wmma<GCN_HID, GCN_OUT><<<blocks, wpb * 32, 0, stream>>>(h1, W2, xw2, N);
    }
    // init output with self-loop term + bias, then scatter edges
    k_selfloop_init<GCN_OUT, true><<<((size_t)N * GCN_OUT + TPB - 1) / TPB, TPB, 0, stream>>>(
        xw2, dinv, b2, out, N);
    k_scatter_edges<GCN_OUT><<<(E + 3) / 4, 128, 0, stream>>>(ei, ew, dinv, xw2, out, E);
}